// ConvLayer_67937792688478
// MI455X (gfx1250) — compile-verified
//
#include <hip/hip_runtime.h>
#include <hip/hip_bf16.h>

typedef __attribute__((ext_vector_type(2))) float v2f;
typedef __attribute__((ext_vector_type(8))) float v8f;

#define FIN 192        // 2A + B input features
#define OCH 256        // ga channels 0..127, gb channels 128..255
#define NSLOT 1024
#define LDS_STRIDE 196 // 196 % 64 == 4 -> 16 rows land on distinct banks
#define BN_EPS 1e-5f

__device__ __forceinline__ float softplus_f(float x) {
    // numerically stable log1p(exp(x))
    return fmaxf(x, 0.0f) + log1pf(expf(-fabsf(x)));
}
__device__ __forceinline__ float sigmoid_f(float x) {
    return 1.0f / (1.0f + expf(-x));
}

// ---------------------------------------------------------------------------
// Kernel 1: gather + concat + dual GEMM via fp32 WMMA, write [NM,256] tile
// buffer, accumulate per-channel sum / sum^2 partials for BatchNorm1.
// One block (256 thr = 8 waves) per atom n; rows are the 16 neighbors.
// ---------------------------------------------------------------------------
__global__ __launch_bounds__(256)
void k_gemm(const float* __restrict__ atomF, const float* __restrict__ nbr,
            const int* __restrict__ idx,
            const float* __restrict__ Wa, const float* __restrict__ Wb,
            float* __restrict__ gemm, float* __restrict__ P1) {
    __shared__ float lds[16 * LDS_STRIDE];
    const int n = blockIdx.x;
    const int t = threadIdx.x;

    // Stage A-tile: 16 rows x 192 floats = 768 float4, 3 per thread.
    #pragma unroll
    for (int i = 0; i < 3; ++i) {
        int q   = t + 256 * i;   // 0..767
        int row = q / 48;        // 0..15  (neighbor slot m)
        int c   = q % 48;        // float4 index within row
        int seg = c >> 4;        // 0: self, 1: gathered neighbor, 2: bond
        int cc  = c & 15;
        const float* src;
        if (seg == 0)      src = atomF + (size_t)n * 64 + cc * 4;
        else if (seg == 1) src = atomF + (size_t)idx[n * 16 + row] * 64 + cc * 4;
        else               src = nbr + ((size_t)n * 16 + row) * 64 + cc * 4;
        float4 v = *(const float4*)src;
        *(float4*)&lds[row * LDS_STRIDE + seg * 64 + cc * 4] = v;
    }
    __syncthreads();

    const int wave  = t >> 5;
    const int lane  = t & 31;
    const int lh    = lane & 15;
    const int khalf = (lane >> 4) << 1;      // 0 or 2 (ISA 16x4 A / 4x16 B striping)
    const int ch0   = wave * 32 + lh;        // two 16-wide output tiles per wave
    const int ch1   = ch0 + 16;
    const float* wr0 = (ch0 < 128 ? Wa + (size_t)ch0 * FIN
                                  : Wb + (size_t)(ch0 - 128) * FIN) + khalf;
    const float* wr1 = (ch1 < 128 ? Wa + (size_t)ch1 * FIN
                                  : Wb + (size_t)(ch1 - 128) * FIN) + khalf;
    const float* arow = &lds[lh * LDS_STRIDE + khalf];

    v8f c0 = {}; v8f c1 = {};
    #pragma unroll
    for (int kc = 0; kc < FIN; kc += 4) {
        v2f a  = *(const v2f*)(arow + kc);   // A[M=lh, K=kc+khalf .. +1]
        v2f b0 = *(const v2f*)(wr0 + kc);    // B[K=kc+khalf .. +1, N=lh]
        v2f b1 = *(const v2f*)(wr1 + kc);
        c0 = __builtin_amdgcn_wmma_f32_16x16x4_f32(false, a, false, b0,
                                                   (short)0, c0, false, false);
        c1 = __builtin_amdgcn_wmma_f32_16x16x4_f32(false, a, false, b1,
                                                   (short)0, c1, false, false);
    }

    // Store D tiles (VGPR v holds row M = v + 8*(lane>>4)) + BN1 stat partials.
    float s0 = 0.f, q0 = 0.f, s1 = 0.f, q1 = 0.f;
    const int mbase = (lane >> 4) * 8;
    size_t rowbase = ((size_t)n * 16 + mbase) * OCH;
    #pragma unroll
    for (int v = 0; v < 8; ++v) {
        float e0 = c0[v], e1 = c1[v];
        gemm[rowbase + (size_t)v * OCH + ch0] = e0;
        gemm[rowbase + (size_t)v * OCH + ch1] = e1;
        s0 += e0; q0 += e0 * e0;
        s1 += e1; q1 += e1 * e1;
    }
    // lanes L and L+16 hold the same channel (rows 0-7 vs 8-15): combine.
    s0 += __shfl_xor(s0, 16, 32);
    q0 += __shfl_xor(q0, 16, 32);
    s1 += __shfl_xor(s1, 16, 32);
    q1 += __shfl_xor(q1, 16, 32);
    if (lane < 16) {
        float* slot = P1 + (size_t)(blockIdx.x & (NSLOT - 1)) * 512;
        atomicAdd(slot + ch0, s0);
        atomicAdd(slot + 256 + ch0, q0);
        atomicAdd(slot + ch1, s1);
        atomicAdd(slot + 256 + ch1, q1);
    }
}

// ---------------------------------------------------------------------------
// Kernel 2: finalize BN1 -> fused per-channel scale/shift (256 channels).
// ---------------------------------------------------------------------------
__global__ __launch_bounds__(256)
void k_fin1(const float* __restrict__ P1,
            const float* __restrict__ g1a, const float* __restrict__ b1a,
            const float* __restrict__ g1b, const float* __restrict__ b1b,
            float* __restrict__ SC1, float cntInv) {
    int ch = threadIdx.x;                 // 0..255
    float s = 0.f, q = 0.f;
    for (int slot = 0; slot < NSLOT; ++slot) {
        s += P1[(size_t)slot * 512 + ch];
        q += P1[(size_t)slot * 512 + 256 + ch];
    }
    float mean = s * cntInv;
    float var  = q * cntInv - mean * mean;
    float rstd = rsqrtf(var + BN_EPS);
    float g = (ch < 128) ? g1a[ch] : g1b[ch - 128];
    float b = (ch < 128) ? b1a[ch] : b1b[ch - 128];
    float scale = g * rstd;
    SC1[ch]       = scale;
    SC1[256 + ch] = b - mean * scale;
}

// ---------------------------------------------------------------------------
// Kernel 3: BN1 + gated activations; bond branch writes out2; atom branch
// sums over m, stores atom_sumed, accumulates BN2 stat partials.
// One block per atom n; thread = (quad 0..3, channel 0..63).
// ---------------------------------------------------------------------------
__global__ __launch_bounds__(256)
void k_act(const float* __restrict__ gemm, const float* __restrict__ nbr,
           const float* __restrict__ SC1,
           float* __restrict__ out2, float* __restrict__ asum,
           float* __restrict__ P2) {
    __shared__ float red[256];
    const int n = blockIdx.x;
    const int t = threadIdx.x;
    const int ch = t & 63;
    const int quad = t >> 6;
    const float saf = SC1[ch],        haf = SC1[256 + ch];
    const float sac = SC1[64 + ch],   hac = SC1[256 + 64 + ch];
    const float sbf = SC1[128 + ch],  hbf = SC1[256 + 128 + ch];
    const float sbc = SC1[192 + ch],  hbc = SC1[256 + 192 + ch];
    float s = 0.f;
    #pragma unroll
    for (int mm = 0; mm < 4; ++mm) {
        size_t g = (size_t)n * 16 + quad * 4 + mm;
        size_t base = g * OCH;
        float af = gemm[base + ch]        * saf + haf;
        float ac = gemm[base + 64 + ch]   * sac + hac;
        s += sigmoid_f(af) * softplus_f(ac);
        float bf = gemm[base + 128 + ch]  * sbf + hbf;
        float bc = gemm[base + 192 + ch]  * sbc + hbc;
        out2[g * 64 + ch] = softplus_f(nbr[g * 64 + ch] +
                                       sigmoid_f(bf) * softplus_f(bc));
    }
    red[t] = s;
    __syncthreads();
    if (quad == 0) {
        float tot = red[ch] + red[64 + ch] + red[128 + ch] + red[192 + ch];
        asum[(size_t)n * 64 + ch] = tot;
        float* slot = P2 + (size_t)(n & (NSLOT - 1)) * 128;
        atomicAdd(slot + ch, tot);
        atomicAdd(slot + 64 + ch, tot * tot);
    }
}

// ---------------------------------------------------------------------------
// Kernel 4: finalize BN2 -> scale/shift (64 channels).
// ---------------------------------------------------------------------------
__global__ __launch_bounds__(64)
void k_fin2(const float* __restrict__ P2, const float* __restrict__ g2a,
            const float* __restrict__ b2a, float* __restrict__ SC2,
            float cntInv) {
    int ch = threadIdx.x;                 // 0..63
    float s = 0.f, q = 0.f;
    for (int slot = 0; slot < NSLOT; ++slot) {
        s += P2[(size_t)slot * 128 + ch];
        q += P2[(size_t)slot * 128 + 64 + ch];
    }
    float mean = s * cntInv;
    float var  = q * cntInv - mean * mean;
    float rstd = rsqrtf(var + BN_EPS);
    float scale = g2a[ch] * rstd;
    SC2[ch]      = scale;
    SC2[64 + ch] = b2a[ch] - mean * scale;
}

// ---------------------------------------------------------------------------
// Kernel 5: out1 = softplus(atom_in + BN2(atom_sumed)).
// ---------------------------------------------------------------------------
__global__ __launch_bounds__(256)
void k_out1(const float* __restrict__ atomF, const float* __restrict__ asum,
            const float* __restrict__ SC2, float* __restrict__ out1,
            int total) {
    int i = blockIdx.x * 256 + threadIdx.x;
    if (i >= total) return;
    int ch = i & 63;
    out1[i] = softplus_f(atomF[i] + asum[i] * SC2[ch] + SC2[64 + ch]);
}

extern "C" void kernel_launch(void* const* d_in, const int* in_sizes, int n_in,
                              void* d_out, int out_size, void* d_ws, size_t ws_size,
                              hipStream_t stream) {
    const float* atomF = (const float*)d_in[0];
    const float* nbr   = (const float*)d_in[1];
    const float* Wa    = (const float*)d_in[2];
    const float* Wb    = (const float*)d_in[3];
    const float* g1a   = (const float*)d_in[4];
    const float* b1a   = (const float*)d_in[5];
    const float* g1b   = (const float*)d_in[6];
    const float* b1b   = (const float*)d_in[7];
    const float* g2a   = (const float*)d_in[8];
    const float* b2a   = (const float*)d_in[9];
    const int*   idx   = (const int*)d_in[10];
    const int N = in_sizes[0] / 64;
    const size_t NM = (size_t)N * 16;

    // Workspace layout (floats)
    float* ws = (float*)d_ws;
    size_t OFF_GEMM = 0;
    size_t OFF_P1   = NM * OCH;                       // [NM,256] intermediate
    size_t OFF_SC1  = OFF_P1 + (size_t)NSLOT * 512;   // BN1 stat partials
    size_t OFF_AS   = OFF_SC1 + 512;                  // BN1 scale/shift
    size_t OFF_P2   = OFF_AS + (size_t)N * 64;        // atom_sumed
    size_t OFF_SC2  = OFF_P2 + (size_t)NSLOT * 128;   // BN2 stat partials

    float* gemm = ws + OFF_GEMM;
    float* P1   = ws + OFF_P1;
    float* SC1  = ws + OFF_SC1;
    float* AS   = ws + OFF_AS;
    float* P2   = ws + OFF_P2;
    float* SC2  = ws + OFF_SC2;

    float* out1 = (float*)d_out;
    float* out2 = out1 + (size_t)N * 64;

    hipMemsetAsync(P1, 0, (size_t)NSLOT * 512 * sizeof(float), stream);
    hipMemsetAsync(P2, 0, (size_t)NSLOT * 128 * sizeof(float), stream);

    k_gemm<<<N, 256, 0, stream>>>(atomF, nbr, idx, Wa, Wb, gemm, P1);
    k_fin1<<<1, 256, 0, stream>>>(P1, g1a, b1a, g1b, b1b, SC1, 1.0f / (float)NM);
    k_act<<<N, 256, 0, stream>>>(gemm, nbr, SC1, out2, AS, P2);
    k_fin2<<<1, 64, 0, stream>>>(P2, g2a, b2a, SC2, 1.0f / (float)N);
    int total = N * 64;
    k_out1<<<(total + 255) / 256, 256, 0, stream>>>(atomF, AS, SC2, out1, total);
}